// SketchRnnLoss_7318624272744
// MI455X (gfx1250) — compile-verified
//
#include <hip/hip_runtime.h>

// Problem constants (from reference)
#define BB   2048
#define NMAX 256
#define NS   200
#define MM   20
#define DD   123   // 6*MM + 3
#define NZ   128

constexpr int NBLK = 1024;
constexpr int NTHR = 256;                     // 8 waves (wave32)
constexpr int WPB  = NTHR / 32;
constexpr int TOTAL_WAVES   = NBLK * WPB;
constexpr int TOTAL_THREADS = NBLK * NTHR;

typedef __attribute__((ext_vector_type(2))) float v2f;
typedef __attribute__((ext_vector_type(8))) float v8f;

__device__ __forceinline__ float fast_exp(float x) {
  return __builtin_amdgcn_exp2f(x * 1.44269504088896340736f);   // v_exp_f32
}
__device__ __forceinline__ float fast_log(float x) {
  return __builtin_amdgcn_logf(x) * 0.69314718055994530942f;    // v_log_f32
}

// ---------------------------------------------------------------------------
// Pass 1: stream everything once, per-block partial sums -> ws
//   ws[0      .. NBLK-1] : partials of (Ls + Lp)
//   ws[NBLK   .. 2NBLK-1]: partials of sum(1 + presig - mu^2 - exp(presig))
// ---------------------------------------------------------------------------
__global__ __launch_bounds__(NTHR) void sketch_main(
    const float* __restrict__ inp, const float* __restrict__ tgt,
    const float* __restrict__ mu,  const float* __restrict__ presig,
    float* __restrict__ ws) {
  const int lane  = threadIdx.x & 31;
  const int wv    = threadIdx.x >> 5;
  const int gwave = blockIdx.x * WPB + wv;
  const int tid   = blockIdx.x * NTHR + threadIdx.x;

  const float EPS    = 1e-6f;
  const float INV2PI = 0.15915494309189533577f;

  float accA = 0.0f;   // Ls + Lp contributions
  float accB = 0.0f;   // raw KL sum

  // ---- Gaussian-mixture rows: one wave per (b, n<NS) row ----
  const int R = BB * NS;
  for (int r = gwave; r < R; r += TOTAL_WAVES) {
    const int b = r / NS;
    const int n = r - b * NS;
    const float* __restrict__ row = inp + ((size_t)b * NMAX + n) * DD;
    const float* __restrict__ tg  = tgt + ((size_t)b * NS   + n) * 5;
    const float x1 = tg[0];
    const float x2 = tg[1];

    float vmix = 0.0f, vlp = 0.0f;
    if (lane < MM) {
      const float* __restrict__ p = row + lane * 6;   // 20 lanes -> 480B contiguous
      const float pi = p[0], m1 = p[1], m2 = p[2];
      const float s1 = p[3], s2 = p[4], cr = p[5];
      const float is1 = __builtin_amdgcn_rcpf(s1);
      const float is2 = __builtin_amdgcn_rcpf(s2);
      const float n1  = (x1 - m1) * is1;
      const float n2  = (x2 - m2) * is2;
      const float omc = 1.0f - cr * cr;
      const float Z   = n1 * n1 + n2 * n2 - 2.0f * cr * n1 * n2;
      const float ex  = fast_exp(-Z * 0.5f * __builtin_amdgcn_rcpf(omc));
      const float dens = ex * INV2PI * is1 * is2 * __builtin_amdgcn_rsqf(omc);
      vmix = pi * dens;
    } else if (lane < MM + 3) {
      const int k   = lane - MM;
      const float q  = row[6 * MM + k];
      const float pv = tg[2 + k];
      vlp = pv * fast_log(q + EPS);
    }
    #pragma unroll
    for (int o = 16; o > 0; o >>= 1) {
      vmix += __shfl_xor(vmix, o, 32);
      vlp  += __shfl_xor(vlp,  o, 32);
    }
    if (lane == 0) accA += fast_log(vmix + EPS) + vlp;
  }

  // ---- pen-state tail rows (n >= NS): p = [0,0,1] -> log(q2 + eps) ----
  const int TAIL = NMAX - NS;          // 56
  const int E = BB * TAIL;
  for (int i = tid; i < E; i += TOTAL_THREADS) {
    const int b = i / TAIL;
    const int n = NS + (i - b * TAIL);
    const float q2 = inp[((size_t)b * NMAX + n) * DD + (DD - 1)];
    accA += fast_log(q2 + EPS);
  }

  // ---- KL term ----
  const int K = BB * NZ;
  for (int i = tid; i < K; i += TOTAL_THREADS) {
    const float ps = presig[i];
    const float m  = mu[i];
    accB += 1.0f + ps - m * m - fast_exp(ps);
  }

  // ---- deterministic block reduction ----
  __shared__ float sA[WPB], sB[WPB];
  #pragma unroll
  for (int o = 16; o > 0; o >>= 1) {
    accA += __shfl_xor(accA, o, 32);
    accB += __shfl_xor(accB, o, 32);
  }
  if (lane == 0) { sA[wv] = accA; sB[wv] = accB; }
  __syncthreads();
  if (threadIdx.x == 0) {
    float a = 0.0f, c = 0.0f;
    #pragma unroll
    for (int i = 0; i < WPB; ++i) { a += sA[i]; c += sB[i]; }
    ws[blockIdx.x]        = a;
    ws[NBLK + blockIdx.x] = c;
  }
}

// ---------------------------------------------------------------------------
// Pass 2: one wave reduces 2x1024 partials with V_WMMA_F32_16X16X4_F32.
// B = all-ones, so D = A x 1 + C accumulates row sums; total of all 1024
// values = sum over the 16 rows of any one column of C.
// ---------------------------------------------------------------------------
__global__ __launch_bounds__(32) void sketch_finalize(
    const float* __restrict__ ws, const float* __restrict__ wkl,
    float* __restrict__ out) {
  const int lane = threadIdx.x;
  const int mrow = lane & 15;
  const int kb   = (lane < 16) ? 0 : 2;   // A 16x4 layout: v0=K{0,2}, v1=K{1,3}

  v8f cA = {};
  v8f cB = {};
  v2f bones = {1.0f, 1.0f};

  #pragma unroll
  for (int chunk = 0; chunk < 16; ++chunk) {
    const float* pa = ws + chunk * 64 + mrow * 4 + kb;
    v2f a;  a.x  = pa[0]; a.y  = pa[1];
    cA = __builtin_amdgcn_wmma_f32_16x16x4_f32(false, a,  false, bones,
                                               (short)0, cA, false, false);
    const float* pb = ws + NBLK + chunk * 64 + mrow * 4 + kb;
    v2f a2; a2.x = pb[0]; a2.y = pb[1];
    cB = __builtin_amdgcn_wmma_f32_16x16x4_f32(false, a2, false, bones,
                                               (short)0, cB, false, false);
  }

  float tA = cA[0]+cA[1]+cA[2]+cA[3]+cA[4]+cA[5]+cA[6]+cA[7]; // rows 0-7 / 8-15
  float tB = cB[0]+cB[1]+cB[2]+cB[3]+cB[4]+cB[5]+cB[6]+cB[7];
  tA += __shfl_xor(tA, 16, 32);   // combine the two half-column sums
  tB += __shfl_xor(tB, 16, 32);

  if (lane == 0) {
    const float Lr  = -tA / ((float)NMAX * (float)BB);
    const float Lkl = -tB / (2.0f * (float)NZ * (float)BB);
    out[0] = Lr + wkl[0] * Lkl;
  }
}

// ---------------------------------------------------------------------------
extern "C" void kernel_launch(void* const* d_in, const int* in_sizes, int n_in,
                              void* d_out, int out_size, void* d_ws, size_t ws_size,
                              hipStream_t stream) {
  (void)in_sizes; (void)n_in; (void)out_size; (void)ws_size;
  const float* inp    = (const float*)d_in[0];
  const float* tgt    = (const float*)d_in[1];
  const float* mu     = (const float*)d_in[2];
  const float* presig = (const float*)d_in[3];
  const float* wkl    = (const float*)d_in[4];
  float* out = (float*)d_out;
  float* ws  = (float*)d_ws;   // needs 2*NBLK floats = 8 KB

  sketch_main<<<NBLK, NTHR, 0, stream>>>(inp, tgt, mu, presig, ws);
  sketch_finalize<<<1, 32, 0, stream>>>(ws, wkl, out);
}